// IMUPreintegrator_53154515255613
// MI455X (gfx1250) — compile-verified
//
#include <hip/hip_runtime.h>
#include <math.h>

typedef float v2f __attribute__((ext_vector_type(2)));
typedef float v8f __attribute__((ext_vector_type(8)));

#define GRAV     9.81007f
#define ANG_COVC 2.87913024e-08f   /* 0.00016968^2 */
#define ACC_COVC 4.0e-06f          /* 0.002^2 */

struct Q4 { float x, y, z, w; };

__device__ __forceinline__ Q4 qmul(Q4 a, Q4 b) {
  Q4 r;
  r.x = a.w*b.x + b.w*a.x + (a.y*b.z - a.z*b.y);
  r.y = a.w*b.y + b.w*a.y + (a.z*b.x - a.x*b.z);
  r.z = a.w*b.z + b.w*a.z + (a.x*b.y - a.y*b.x);
  r.w = a.w*b.w - (a.x*b.x + a.y*b.y + a.z*b.z);
  return r;
}
__device__ __forceinline__ Q4 qnorm(Q4 a) {
  float n = rsqrtf(a.x*a.x + a.y*a.y + a.z*a.z + a.w*a.w);
  Q4 r = { a.x*n, a.y*n, a.z*n, a.w*n };
  return r;
}
__device__ __forceinline__ void qrot(Q4 q, float vx, float vy, float vz,
                                     float* ox, float* oy, float* oz) {
  float tx = 2.f*(q.y*vz - q.z*vy);
  float ty = 2.f*(q.z*vx - q.x*vz);
  float tz = 2.f*(q.x*vy - q.y*vx);
  *ox = vx + q.w*tx + (q.y*tz - q.z*ty);
  *oy = vy + q.w*ty + (q.z*tx - q.x*tz);
  *oz = vz + q.w*tz + (q.x*ty - q.y*tx);
}
__device__ __forceinline__ Q4 so3exp(float px, float py, float pz) {
  float t2 = px*px + py*py + pz*pz;
  float k, w;
  if (t2 < 1e-8f) { k = 0.5f - t2*(1.f/48.f); w = 1.f - t2*0.125f; }
  else { float t = sqrtf(t2); k = __sinf(0.5f*t)/t; w = __cosf(0.5f*t); }
  Q4 r = { k*px, k*py, k*pz, w };
  return r;
}
__device__ __forceinline__ void qmat(Q4 q, float R[3][3]) {
  float x=q.x, y=q.y, z=q.z, w=q.w;
  R[0][0]=1.f-2.f*(y*y+z*z); R[0][1]=2.f*(x*y-z*w);     R[0][2]=2.f*(x*z+y*w);
  R[1][0]=2.f*(x*y+z*w);     R[1][1]=1.f-2.f*(x*x+z*z); R[1][2]=2.f*(y*z-x*w);
  R[2][0]=2.f*(x*z-y*w);     R[2][1]=2.f*(y*z+x*w);     R[2][2]=1.f-2.f*(x*x+y*y);
}
__device__ __forceinline__ void rightjac(float px, float py, float pz, float J[3][3]) {
  float t2 = px*px + py*py + pz*pz;
  float c1, c2;
  if (t2 < 1e-8f) { c1 = 0.5f - t2*(1.f/24.f); c2 = (1.f/6.f) - t2*(1.f/120.f); }
  else { float t = sqrtf(t2); c1 = (1.f - __cosf(t))/t2; c2 = (t - __sinf(t))/(t2*t); }
  float K[3][3]  = {{0.f,-pz,py},{pz,0.f,-px},{-py,px,0.f}};
  float K2[3][3] = {{-(py*py+pz*pz), px*py, px*pz},
                    {px*py, -(px*px+pz*pz), py*pz},
                    {px*pz, py*pz, -(px*px+py*py)}};
#pragma unroll
  for (int i=0;i<3;i++)
#pragma unroll
    for (int j=0;j<3;j++)
      J[i][j] = (i==j?1.f:0.f) - c1*K[i][j] + c2*K2[i][j];
}

/* ================= Kernel 1: scans (rot/vel/pos) ====================== */
#define K1T 256
#define CMAX 8

__global__ __launch_bounds__(K1T)
void k1_scan(const float* __restrict__ dt, const float* __restrict__ ang,
             const float* __restrict__ acc, const float* __restrict__ init_pos,
             const float* __restrict__ init_rot, const float* __restrict__ init_vel,
             float* __restrict__ out_rot, float* __restrict__ out_vel,
             float* __restrict__ out_pos, float* __restrict__ wsq, int F) {
  const int b = blockIdx.x, tid = threadIdx.x;
  int C = F / K1T; if (C > CMAX) C = CMAX;
  const int s0 = tid * C;
  __shared__ float4 sbuf[K1T];

  Q4 lq[CMAX]; float ldt[CMAX];
  Q4 acq = {0.f,0.f,0.f,1.f};
#pragma unroll
  for (int k=0;k<CMAX;k++){
    if (k>=C) break;
    int s = s0+k;
    float d = dt[(size_t)b*F+s];
    ldt[k]=d;
    size_t ai = ((size_t)b*F+s)*3;
    __builtin_prefetch(&acc[ai], 0, 0);
    Q4 dq = so3exp(ang[ai]*d, ang[ai+1]*d, ang[ai+2]*d);
    acq = qmul(acq, dq);
    lq[k] = acq;
  }
  /* block-wide inclusive scan of chunk quaternions */
  float4 v = make_float4(acq.x, acq.y, acq.z, acq.w);
  sbuf[tid] = v; __syncthreads();
  for (int off=1; off<K1T; off<<=1){
    float4 o = make_float4(0.f,0.f,0.f,1.f);
    if (tid >= off) o = sbuf[tid-off];
    __syncthreads();
    if (tid >= off){
      Q4 qa={o.x,o.y,o.z,o.w}, qb={v.x,v.y,v.z,v.w};
      Q4 m = qmul(qa,qb);
      v = make_float4(m.x,m.y,m.z,m.w);
    }
    sbuf[tid] = v; __syncthreads();
  }
  Q4 E = {0.f,0.f,0.f,1.f};
  if (tid>0){ float4 e = sbuf[tid-1]; E.x=e.x; E.y=e.y; E.z=e.z; E.w=e.w; }

  Q4 qr0 = { init_rot[b*4+0], init_rot[b*4+1], init_rot[b*4+2], init_rot[b*4+3] };
  float ip0=init_pos[b*3+0], ip1=init_pos[b*3+1], ip2=init_pos[b*3+2];
  float iv0=init_vel[b*3+0], iv1=init_vel[b*3+1], iv2=init_vel[b*3+2];
  if (tid==0){
    size_t qi=(size_t)b*(F+1)*4;
    wsq[qi+0]=0.f; wsq[qi+1]=0.f; wsq[qi+2]=0.f; wsq[qi+3]=1.f;
  }

  float rax[CMAX],ray[CMAX],raz[CMAX];
  float lvx[CMAX],lvy[CMAX],lvz[CMAX],lts[CMAX];
  Q4 pq = qnorm(E);
  float svx=0.f,svy=0.f,svz=0.f,st=0.f;
#pragma unroll
  for (int k=0;k<CMAX;k++){
    if (k>=C) break;
    int s = s0+k; float d = ldt[k];
    Q4 iq = qnorm(qmul(E, lq[k]));
    size_t qi = ((size_t)b*(F+1)+s+1)*4;
    wsq[qi+0]=iq.x; wsq[qi+1]=iq.y; wsq[qi+2]=iq.z; wsq[qi+3]=iq.w;
    Q4 rq = qmul(qr0, iq);
    size_t ri = ((size_t)b*F+s)*4;
    out_rot[ri+0]=rq.x; out_rot[ri+1]=rq.y; out_rot[ri+2]=rq.z; out_rot[ri+3]=rq.w;
    /* a = acc - R(rot)^-1 g */
    Q4 rqi = { -rq.x, -rq.y, -rq.z, rq.w };
    float gx,gy,gz; qrot(rqi, 0.f, 0.f, GRAV, &gx,&gy,&gz);
    size_t ai = ((size_t)b*F+s)*3;
    float a0 = acc[ai+0]-gx, a1 = acc[ai+1]-gy, a2 = acc[ai+2]-gz;
    float r0,r1,r2; qrot(pq, a0,a1,a2, &r0,&r1,&r2);   /* ra = R(incre_r[s]) a */
    rax[k]=r0; ray[k]=r1; raz[k]=r2;
    svx += r0*d; svy += r1*d; svz += r2*d; st += d;
    lvx[k]=svx; lvy[k]=svy; lvz[k]=svz; lts[k]=st;
    pq = iq;
  }
  /* block-wide additive scan of (sum ra*dt, sum dt) */
  v = make_float4(svx,svy,svz,st);
  sbuf[tid] = v; __syncthreads();
  for (int off=1; off<K1T; off<<=1){
    float4 o = make_float4(0.f,0.f,0.f,0.f);
    if (tid >= off) o = sbuf[tid-off];
    __syncthreads();
    if (tid >= off){ v.x+=o.x; v.y+=o.y; v.z+=o.z; v.w+=o.w; }
    sbuf[tid] = v; __syncthreads();
  }
  float ovx=0.f,ovy=0.f,ovz=0.f,ot=0.f;
  if (tid>0){ float4 e = sbuf[tid-1]; ovx=e.x; ovy=e.y; ovz=e.z; ot=e.w; }

  float ldx[CMAX],ldy[CMAX],ldz[CMAX];
  float sdx=0.f,sdy=0.f,sdz=0.f;
#pragma unroll
  for (int k=0;k<CMAX;k++){
    if (k>=C) break;
    int s = s0+k; float d = ldt[k];
    float vex = ovx + (k? lvx[k-1]:0.f);
    float vey = ovy + (k? lvy[k-1]:0.f);
    float vez = ovz + (k? lvz[k-1]:0.f);
    float vix = ovx + lvx[k], viy = ovy + lvy[k], viz = ovz + lvz[k];
    float w0,w1,w2; qrot(qr0, vix,viy,viz, &w0,&w1,&w2);
    size_t vi = ((size_t)b*F+s)*3;
    out_vel[vi+0]=iv0+w0; out_vel[vi+1]=iv1+w1; out_vel[vi+2]=iv2+w2;
    sdx += vex*d + 0.5f*rax[k]*d*d;
    sdy += vey*d + 0.5f*ray[k]*d*d;
    sdz += vez*d + 0.5f*raz[k]*d*d;
    ldx[k]=sdx; ldy[k]=sdy; ldz[k]=sdz;
  }
  __syncthreads();
  v = make_float4(sdx,sdy,sdz,0.f);
  sbuf[tid] = v; __syncthreads();
  for (int off=1; off<K1T; off<<=1){
    float4 o = make_float4(0.f,0.f,0.f,0.f);
    if (tid >= off) o = sbuf[tid-off];
    __syncthreads();
    if (tid >= off){ v.x+=o.x; v.y+=o.y; v.z+=o.z; }
    sbuf[tid] = v; __syncthreads();
  }
  float odx=0.f,ody=0.f,odz=0.f;
  if (tid>0){ float4 e = sbuf[tid-1]; odx=e.x; ody=e.y; odz=e.z; }
#pragma unroll
  for (int k=0;k<CMAX;k++){
    if (k>=C) break;
    int s = s0+k;
    float px = odx + ldx[k], py = ody + ldy[k], pz = odz + ldz[k];
    float ti = ot + lts[k];
    float w0,w1,w2; qrot(qr0, px,py,pz, &w0,&w1,&w2);
    size_t pi = ((size_t)b*F+s)*3;
    out_pos[pi+0]=ip0+w0+iv0*ti; out_pos[pi+1]=ip1+w1+iv1*ti; out_pos[pi+2]=ip2+w2+iv2*ti;
  }
}

/* ============ Kernel 2: covariance via WMMA 16x16x4 f32 =============== */
#define NW 4
#define TS 17
#define TFLOATS (16*TS)

__device__ __forceinline__ void lds_fence_wave() {
  asm volatile("s_wait_dscnt 0" ::: "memory");
}
__device__ __forceinline__ v8f wmma4(v2f a, v2f b, v8f c) {
  return __builtin_amdgcn_wmma_f32_16x16x4_f32(false, a, false, b, (short)0, c, false, false);
}
/* 16x16 (padded 9x9) matmul: C += Ta * Tb  (or Ta * Tb^T), tiles in LDS */
__device__ __forceinline__ v8f mm16(const float* Ta, const float* Tb, v8f c,
                                    int lane, bool transB) {
  int m = lane & 15;
  int koff = (lane >> 4) << 1;
#pragma unroll
  for (int kt=0; kt<4; kt++){
    int k0 = kt*4 + koff;
    v2f a; a.x = Ta[m*TS + k0]; a.y = Ta[m*TS + k0 + 1];
    v2f b;
    if (!transB){ b.x = Tb[k0*TS + m];  b.y = Tb[(k0+1)*TS + m]; }
    else        { b.x = Tb[m*TS + k0];  b.y = Tb[m*TS + k0 + 1]; }
    c = wmma4(a, b, c);
  }
  return c;
}
__device__ __forceinline__ void store_c(float* T, v8f c, int lane) {
  int n = lane & 15; int mb = (lane>>4)*8;
#pragma unroll
  for (int r=0;r<8;r++) T[(mb+r)*TS + n] = c[r];
}
__device__ __forceinline__ void set_id(float* T, int lane) {
  for (int idx=lane; idx<256; idx+=32){
    int r=idx>>4, c=idx&15;
    T[r*TS+c] = (r==c && r<9) ? 1.f : 0.f;
  }
}
__device__ __forceinline__ void copy_tile(float* D, const float* S, int lane) {
  for (int idx=lane; idx<256; idx+=32){
    int r=idx>>4, c=idx&15;
    D[r*TS+c] = S[r*TS+c];
  }
}
__device__ __forceinline__ Q4 load_q(const float* wsq, size_t i) {
  Q4 q = { wsq[i*4+0], wsq[i*4+1], wsq[i*4+2], wsq[i*4+3] };
  return q;
}
/* A[t] (9x9 padded): rows 0:3 = drT, 3:6 = -M dt, 6:9 = [-0.5 M dt^2, I dt, I] */
__device__ __forceinline__ void build_A(float* T, int lane, int b, int F, int t,
                                        const float* dt, const float* ang,
                                        const float* acc, const float* wsq) {
  if (t >= F) { set_id(T, lane); return; }
  size_t si = (size_t)b*F + t;
  float d = dt[si];
  float drm[3][3]; { Q4 dq = so3exp(ang[si*3]*d, ang[si*3+1]*d, ang[si*3+2]*d); qmat(dq, drm); }
  float ir[3][3];  { Q4 iq = load_q(wsq, (size_t)b*(F+1)+t+1); qmat(iq, ir); }
  float a0=acc[si*3], a1=acc[si*3+1], a2=acc[si*3+2];
  float Ha[3][3] = {{0.f,-a2,a1},{a2,0.f,-a0},{-a1,a0,0.f}};
  float M[3][3];
#pragma unroll
  for (int i=0;i<3;i++)
#pragma unroll
    for (int j=0;j<3;j++)
      M[i][j] = ir[i][0]*Ha[0][j] + ir[i][1]*Ha[1][j] + ir[i][2]*Ha[2][j];
  for (int idx=lane; idx<256; idx+=32){
    int r=idx>>4, c=idx&15; float v=0.f;
    if (r<9 && c<9){
      if (r<3 && c<3)              v = drm[c][r];
      else if (r<6 && c<3)         v = -M[r-3][c]*d;
      else if (r>=6 && c<3)        v = -0.5f*M[r-6][c]*d*d;
      else if (r>=6 && c>=3 && c<6) v = (r-6==c-3) ? d : 0.f;
      else                          v = (r==c) ? 1.f : 0.f;
    }
    T[r*TS+c] = v;
  }
}
/* Bc at step j=t-1: (ANG*Bg Bg^T + ACC*Ba Ba^T)/dt, Bg=[Jr dt;0;0], Ba=[0;ir dt;0.5 dr dt^2] */
__device__ __forceinline__ void build_Bc(float* T, int lane, int b, int F, int t,
                                         const float* dt, const float* ang,
                                         const float* wsq) {
  int j = t-1;
  size_t si = (size_t)b*F + j;
  float d = dt[si];
  float px = ang[si*3]*d, py = ang[si*3+1]*d, pz = ang[si*3+2]*d;
  float Jr[3][3]; rightjac(px,py,pz,Jr);
  float drm[3][3]; { Q4 dq = so3exp(px,py,pz); qmat(dq, drm); }
  float ir[3][3];  { Q4 iq = load_q(wsq, (size_t)b*(F+1)+t); qmat(iq, ir); }
  float invd = 1.f/d;
  for (int idx=lane; idx<256; idx+=32){
    int r=idx>>4, c=idx&15; float v=0.f;
    if (r<9 && c<9){
      float s = 0.f;
      if (r<3 && c<3){
        s = ANG_COVC * d * d *
            (Jr[r][0]*Jr[c][0] + Jr[r][1]*Jr[c][1] + Jr[r][2]*Jr[c][2]);
      } else if (r>=3 && c>=3){
        float br[3], bc[3];
#pragma unroll
        for (int i=0;i<3;i++){
          br[i] = (r<6) ? ir[r-3][i]*d : 0.5f*drm[r-6][i]*d*d;
          bc[i] = (c<6) ? ir[c-3][i]*d : 0.5f*drm[c-6][i]*d*d;
        }
        s = ACC_COVC * (br[0]*bc[0] + br[1]*bc[1] + br[2]*bc[2]);
      }
      v = s * invd;
    }
    T[r*TS+c] = v;
  }
}

__global__ __launch_bounds__(NW*32)
void k2_cov(const float* __restrict__ dt, const float* __restrict__ ang,
            const float* __restrict__ acc, const float* __restrict__ wsq,
            float* __restrict__ out_cov, int F) {
  const int b = blockIdx.x;
  const int wave = threadIdx.x >> 5;
  const int lane = threadIdx.x & 31;
  const int L = F / NW;
  const int t0 = 1 + wave*L, t1 = (wave+1)*L;

  __shared__ float tiles[NW][4][TFLOATS];
  __shared__ float shA[NW][TFLOATS];   /* chunk products, later per-wave S */
  __shared__ float shR[NW][TFLOATS];   /* chunk suffix products */
  float* TA = tiles[wave][0];
  float* TG = tiles[wave][1];
  float* TB = tiles[wave][2];
  float* TT = tiles[wave][3];

  /* Phase A: per-wave chunk product C_w = A[t0]..A[t1] */
  set_id(TG, lane); lds_fence_wave();
  for (int t=t0; t<=t1; t++){
    build_A(TA, lane, b, F, t, dt, ang, acc, wsq);
    lds_fence_wave();
    v8f g = {0.f,0.f,0.f,0.f,0.f,0.f,0.f,0.f};
    g = mm16(TG, TA, g, lane, false);       /* P <- P * A[t] */
    store_c(TG, g, lane);
    lds_fence_wave();
  }
  copy_tile(shA[wave], TG, lane);
  __syncthreads();

  /* Phase B: wave 0 computes suffix products R_w = C_{w+1}..C_{NW-1} */
  if (wave==0){
    set_id(shR[NW-1], lane); lds_fence_wave();
    for (int w2=NW-2; w2>=0; w2--){
      v8f r = {0.f,0.f,0.f,0.f,0.f,0.f,0.f,0.f};
      r = mm16(shA[w2+1], shR[w2+1], r, lane, false);
      store_c(shR[w2], r, lane);
      lds_fence_wave();
    }
  }
  __syncthreads();

  /* Phase C: replay chunk descending, accumulate S += G Bc G^T with G = A[t]..A[F] */
  copy_tile(TG, shR[wave], lane);
  lds_fence_wave();
  v8f sacc = {0.f,0.f,0.f,0.f,0.f,0.f,0.f,0.f};
  for (int t=t1; t>=t0; t--){
    build_A(TA, lane, b, F, t, dt, ang, acc, wsq);
    lds_fence_wave();
    v8f g = {0.f,0.f,0.f,0.f,0.f,0.f,0.f,0.f};
    g = mm16(TA, TG, g, lane, false);       /* G <- A[t] * G */
    store_c(TG, g, lane);
    lds_fence_wave();
    build_Bc(TB, lane, b, F, t, dt, ang, wsq);
    lds_fence_wave();
    v8f t1v = {0.f,0.f,0.f,0.f,0.f,0.f,0.f,0.f};
    t1v = mm16(TG, TB, t1v, lane, false);   /* T1 = G * Bc */
    store_c(TT, t1v, lane);
    lds_fence_wave();
    sacc = mm16(TT, TG, sacc, lane, true);  /* S += T1 * G^T */
  }
  store_c(shA[wave], sacc, lane);
  lds_fence_wave();
  __syncthreads();

  /* Reduce across waves, write 9x9 cov */
  int ti = threadIdx.x;
  if (ti < 81){
    int r = ti/9, c = ti%9;
    float s = 0.f;
#pragma unroll
    for (int w=0; w<NW; w++) s += shA[w][r*TS+c];
    out_cov[(size_t)b*81 + r*9 + c] = s;
  }
}

/* ========================= launch ===================================== */
extern "C" void kernel_launch(void* const* d_in, const int* in_sizes, int n_in,
                              void* d_out, int out_size, void* d_ws, size_t ws_size,
                              hipStream_t stream) {
  const float* dt       = (const float*)d_in[0];
  const float* ang      = (const float*)d_in[1];
  const float* acc      = (const float*)d_in[2];
  const float* init_pos = (const float*)d_in[3];
  const float* init_rot = (const float*)d_in[4];
  const float* init_vel = (const float*)d_in[5];
  const int B = in_sizes[3] / 3;          /* init_pos is (B,1,3) */
  const int F = in_sizes[0] / B;          /* dt is (B,F,1) */

  float* out     = (float*)d_out;
  float* out_rot = out;
  float* out_vel = out_rot + (size_t)B*F*4;
  float* out_pos = out_vel + (size_t)B*F*3;
  float* out_cov = out_pos + (size_t)B*F*3;
  float* wsq     = (float*)d_ws;          /* B*(F+1)*4 floats of prefix quats */

  k1_scan<<<B, K1T, 0, stream>>>(dt, ang, acc, init_pos, init_rot, init_vel,
                                 out_rot, out_vel, out_pos, wsq, F);
  k2_cov<<<B, NW*32, 0, stream>>>(dt, ang, acc, wsq, out_cov, F);
}